// VectorQuantizerEMA_88759794139424
// MI455X (gfx1250) — compile-verified
//
#include <hip/hip_runtime.h>
#include <hip/hip_bf16.h>
#include <math.h>

typedef float v2f __attribute__((ext_vector_type(2)));
typedef float v8f __attribute__((ext_vector_type(8)));

#define KCODES 512
#define DDIM   64
#define HW     4096
#define ROWS   128            // rows per block
#define NTOT   131072         // 32*4096 vectors
#define NUMEL  8388608        // 32*64*4096 elements
#define GAMMA_ 0.99f
#define BETA_  0.25f
#define EPS_   1e-5f

// output layout (floats), tuple concatenated flat in return order
#define O_QUANT 0
#define O_LOSS  8388608
#define O_PERP  8388609
#define O_EMB   8388610
#define O_SCS   8421378
#define O_EMAW  8421890

// dynamic LDS layout (float units)
#define EB_STRIDE 65
#define XS_STRIDE 65
#define EB_OFF  0                        // 512*65 = 33280
#define XS_OFF  (512*65)                 // 128*65 = 8320
#define CN_OFF  (XS_OFF + 128*65)        // 512
#define IDX_OFF (CN_OFF + 512)           // 128 (uint)
#define LC_OFF  (IDX_OFF + 128)          // 512 (uint)
#define WSSE_OFF (LC_OFF + 512)          // 8
#define LDS_FLOATS (WSSE_OFF + 8)

__global__ __launch_bounds__(256) void vq_zero(float* p, int n) {
    int i = blockIdx.x * 256 + threadIdx.x;
    if (i < n) p[i] = 0.0f;
}

__global__ __launch_bounds__(256) void vq_main(const float* __restrict__ x,
                                               const float* __restrict__ emb,
                                               float* __restrict__ quant,
                                               float* __restrict__ g_counts,
                                               float* __restrict__ g_dw,
                                               float* __restrict__ g_sseP) {
    extern __shared__ float smem[];
    float*    eb   = smem + EB_OFF;
    float*    xs   = smem + XS_OFF;
    float*    cn   = smem + CN_OFF;
    unsigned* widx = (unsigned*)(smem + IDX_OFF);
    unsigned* lcnt = (unsigned*)(smem + LC_OFF);
    float*    wsse = smem + WSSE_OFF;

    const int tid = threadIdx.x;
    const int b   = blockIdx.x >> 5;                 // 32 blocks per batch image
    const int hw0 = (blockIdx.x & 31) * ROWS;
    const float* xBase = x + (size_t)b * DDIM * HW + hw0;

    // stage codebook: emb[c][d] -> eb[c*65+d]  (conflict-free scalar stores)
    for (int it = 0; it < 128; ++it) {
        int idx = it * 256 + tid;                    // 32768
        int c = idx >> 6, d = idx & 63;
        eb[c * EB_STRIDE + d] = emb[idx];
    }
    // stage x tile transposed: x[b][d][hw0+i] -> xs[i*65+d]
    for (int it = 0; it < 32; ++it) {
        int idx = it * 256 + tid;                    // 8192
        int d = idx >> 7, i = idx & 127;
        xs[i * XS_STRIDE + d] = xBase[(size_t)d * HW + i];
    }
    lcnt[tid] = 0u; lcnt[tid + 256] = 0u;
    __syncthreads();

    // per-code squared norms
    for (int c = tid; c < KCODES; c += 256) {
        float s = 0.0f;
        for (int d = 0; d < DDIM; ++d) { float e = eb[c * EB_STRIDE + d]; s += e * e; }
        cn[c] = s;
    }
    __syncthreads();

    const int lane  = tid & 31;
    const int wave  = tid >> 5;
    const int nlane = lane & 15;
    const int hi    = lane >> 4;        // half-wave
    const int kb    = hi * 2;           // K sub-offset per ISA A/B layout
    const int i0    = wave * 16;        // first row of this wave
    const int row   = i0 + nlane;

    // A operand cached in registers: 16 chunks of K=4
    v2f a[16];
#pragma unroll
    for (int c16 = 0; c16 < 16; ++c16) {
        a[c16].x = xs[row * XS_STRIDE + c16 * 4 + kb];
        a[c16].y = xs[row * XS_STRIDE + c16 * 4 + kb + 1];
    }

    float minV[8];
    int   minI[8];
#pragma unroll
    for (int v = 0; v < 8; ++v) { minV[v] = 3.4e38f; minI[v] = 0; }

    for (int t = 0; t < 32; ++t) {                    // 32 tiles of 16 codes
        v8f acc = {0.f, 0.f, 0.f, 0.f, 0.f, 0.f, 0.f, 0.f};
        const int code = t * 16 + nlane;              // this lane's column code
        const float* ebc = eb + code * EB_STRIDE;
#pragma unroll
        for (int c16 = 0; c16 < 16; ++c16) {          // K = 64 in chunks of 4
            v2f bv;
            bv.x = ebc[c16 * 4 + kb];
            bv.y = ebc[c16 * 4 + kb + 1];
            acc = __builtin_amdgcn_wmma_f32_16x16x4_f32(
                false, a[c16], false, bv, (short)0, acc, false, false);
        }
        const float cnv = cn[code];
#pragma unroll
        for (int v = 0; v < 8; ++v) {
            float cand = cnv - 2.0f * acc[v];         // ||e||^2 - 2 x.e (row norm constant)
            if (cand < minV[v]) { minV[v] = cand; minI[v] = code; }
        }
    }

    // reduce argmin across the 16 lanes holding each row (xor 1..8 stays in half)
#pragma unroll
    for (int v = 0; v < 8; ++v) {
        float mv = minV[v]; int mi = minI[v];
#pragma unroll
        for (int m = 1; m <= 8; m <<= 1) {
            float ov = __shfl_xor(mv, m, 32);
            int   oi = __shfl_xor(mi, m, 32);
            if (ov < mv || (ov == mv && oi < mi)) { mv = ov; mi = oi; }
        }
        minV[v] = mv; minI[v] = mi;
    }
    if (nlane == 0) {
        int base = i0 + hi * 8;                       // VGPR v: rows v / v+8
#pragma unroll
        for (int v = 0; v < 8; ++v) widx[base + v] = (unsigned)minI[v];
    }
    __syncthreads();

    // counts: LDS histogram then global float atomics
    if (tid < ROWS) atomicAdd(&lcnt[widx[tid]], 1u);
    __syncthreads();
    for (int c = tid; c < KCODES; c += 256) {
        unsigned cv = lcnt[c];
        if (cv) atomicAdd(&g_counts[c], (float)cv);
    }

    // dw scatter: dw[code][d] += flat[row][d]
    for (int it = 0; it < 32; ++it) {
        int idx = it * 256 + tid;                     // 8192
        int r = idx >> 6, d = idx & 63;
        atomicAdd(&g_dw[widx[r] * DDIM + d], xs[r * XS_STRIDE + d]);
    }

    // quant gather (quant_st == quant numerically) + SSE partial
    float sse = 0.0f;
    float* qBase = quant + (size_t)b * DDIM * HW + hw0;
    for (int it = 0; it < 32; ++it) {
        int idx = it * 256 + tid;                     // 8192
        int d = idx >> 7, i = idx & 127;
        float q  = eb[widx[i] * EB_STRIDE + d];
        float xv = xs[i * XS_STRIDE + d];
        qBase[(size_t)d * HW + i] = q;
        float df = q - xv;
        sse += df * df;
    }
#pragma unroll
    for (int m = 1; m < 32; m <<= 1) sse += __shfl_xor(sse, m, 32);
    if (lane == 0) wsse[wave] = sse;
    __syncthreads();
    if (tid == 0) {                                   // deterministic per-block partial
        float s = 0.0f;
        for (int w = 0; w < 8; ++w) s += wsse[w];
        g_sseP[blockIdx.x] = s;
    }
}

__global__ __launch_bounds__(512) void vq_finalize(const float* __restrict__ ema_cs,
                                                   const float* __restrict__ ema_w,
                                                   const float* __restrict__ g_counts,
                                                   const float* __restrict__ g_dw,
                                                   const float* __restrict__ g_sseP,
                                                   float* __restrict__ out) {
    __shared__ float red[512];
    const int tid = threadIdx.x;

    const float counts = g_counts[tid];
    const float new_cs = GAMMA_ * ema_cs[tid] + (1.0f - GAMMA_) * counts;

    // n = sum(new_cs)
    red[tid] = new_cs;
    __syncthreads();
    for (int s = 256; s > 0; s >>= 1) { if (tid < s) red[tid] += red[tid + s]; __syncthreads(); }
    const float n = red[0];
    __syncthreads();

    // sum p*log p
    const float p  = counts / (float)NTOT;
    red[tid] = (p > 0.0f) ? p * logf(p) : 0.0f;
    __syncthreads();
    for (int s = 256; s > 0; s >>= 1) { if (tid < s) red[tid] += red[tid + s]; __syncthreads(); }
    const float plogp = red[0];
    __syncthreads();

    // total SSE over 1024 block partials (deterministic tree)
    red[tid] = g_sseP[tid] + g_sseP[tid + 512];
    __syncthreads();
    for (int s = 256; s > 0; s >>= 1) { if (tid < s) red[tid] += red[tid + s]; __syncthreads(); }
    const float sseTot = red[0];

    const float smoothed = (new_cs + EPS_) / (n + (float)KCODES * EPS_) * n;
    out[O_SCS + tid] = smoothed;

    for (int d = 0; d < DDIM; ++d) {
        int idx = tid * DDIM + d;
        float w = GAMMA_ * ema_w[idx] + (1.0f - GAMMA_) * g_dw[idx];
        out[O_EMAW + idx] = w;
        out[O_EMB + idx]  = w / smoothed;
    }
    if (tid == 0) {
        out[O_LOSS] = BETA_ * sseTot / (float)NUMEL;
        out[O_PERP] = expf(-plogp);
    }
}

extern "C" void kernel_launch(void* const* d_in, const int* in_sizes, int n_in,
                              void* d_out, int out_size, void* d_ws, size_t ws_size,
                              hipStream_t stream) {
    const float* x      = (const float*)d_in[0];   // [32,64,64,64]
    const float* emb    = (const float*)d_in[1];   // [512,64]
    const float* ema_cs = (const float*)d_in[2];   // [512]
    const float* ema_w  = (const float*)d_in[3];   // [512,64]
    float* out = (float*)d_out;
    float* ws  = (float*)d_ws;

    float* g_counts = ws;                  // 512
    float* g_dw     = ws + 512;            // 32768
    float* g_sseP   = ws + 512 + 32768;    // 1024

    // zero counts + dw every call (ws is not re-poisoned between replays)
    vq_zero<<<(512 + 32768 + 255) / 256, 256, 0, stream>>>(ws, 512 + 32768);

    const size_t ldsBytes = (size_t)LDS_FLOATS * sizeof(float);
    vq_main<<<1024, 256, ldsBytes, stream>>>(x, emb, out + O_QUANT, g_counts, g_dw, g_sseP);

    vq_finalize<<<1, 512, 0, stream>>>(ema_cs, ema_w, g_counts, g_dw, g_sseP, out);
}